// SmoeDecoder_61942018343168
// MI455X (gfx1250) — compile-verified
//
#include <hip/hip_runtime.h>

// SmoE decoder on gfx1250 (MI455X).
// One wave32 per batch element; the 16x16 output grid maps exactly onto the
// V_WMMA_F32_16X16X4_F32 C/D fragment. Per gaussian k, the exponent tile
//   E[i][j] = -0.5*((a00*dx_j + a10*dy_i)^2 + (a11*dy_i)^2)
// is rank-3 in (row i, col j), so one 16x16x4 f32 WMMA produces all 256
// exponents. 4 WMMAs per batch, then v_exp_f32, normalize, clamp, store.

typedef __attribute__((ext_vector_type(2))) float v2f;
typedef __attribute__((ext_vector_type(8))) float v8f;

#define BSZ 16
#define KG 4
#define WAVES_PER_BLOCK 8

__global__ __launch_bounds__(32 * WAVES_PER_BLOCK)
void smoe_decoder_wmma(const float* __restrict__ params,
                       float* __restrict__ out,
                       int nBatch) {
  const int lane = threadIdx.x & 31;
  // Force wave-uniform batch index -> params are fetched with s_load.
  const int wave = __builtin_amdgcn_readfirstlane((int)(threadIdx.x >> 5));
  const int b = blockIdx.x * WAVES_PER_BLOCK + wave;
  if (b >= nBatch) return;  // uniform per wave; EXEC stays all-1s inside

  const float* __restrict__ p = params + (size_t)b * (3 * KG + 4 * KG);

  const int half = lane >> 4;       // 0: K/M low half, 1: high half
  const int r    = lane & 15;       // row (for A) / col (for B) / N (for D)
  const float gv = (float)r * (1.0f / 15.0f);  // linspace(0,1,16)[r]

  v8f nsum = {};  // sum_k exp(E_k)
  v8f wsum = {};  // sum_k nue_k * exp(E_k)

#pragma unroll
  for (int k = 0; k < KG; ++k) {
    const float cx  = p[k];
    const float cy  = p[KG + k];
    const float nue = p[2 * KG + k];
    const float a00 = p[3 * KG + 4 * k + 0];
    const float a10 = p[3 * KG + 4 * k + 2];
    const float a11 = p[3 * KG + 4 * k + 3];

    // Column terms (gx = gv at col r) and row terms (gy = gv at row r).
    const float pj = a00 * (gv - cx);   // a00*(gx_j - cx)
    const float qi = a10 * (gv - cy);   // a10*(gy_i - cy)
    const float ri = a11 * (gv - cy);   // a11*(gy_i - cy)

    // A (16x4, M=r rows):  row_i = [1, -q_i, -0.5*(q_i^2+r_i^2), 0]
    //   lanes 0-15 hold K=0,1; lanes 16-31 hold K=2,3 (ISA A-layout).
    v2f afrag;
    afrag.x = half ? (-0.5f * (qi * qi + ri * ri)) : 1.0f;
    afrag.y = half ? 0.0f : (-qi);

    // B (4x16, N=r cols):  col_j = [-0.5*p_j^2, p_j, 1, 0]^T
    //   VGPR g / lane-half holds row K = g + 2*half (mirrors C/D layout).
    v2f bfrag;
    bfrag.x = half ? 1.0f : (-0.5f * pj * pj);
    bfrag.y = half ? 0.0f : pj;

    v8f cz = {};
    // (neg_a, A, neg_b, B, c_mod, C, reuse_a, reuse_b)
    v8f e = __builtin_amdgcn_wmma_f32_16x16x4_f32(
        false, afrag, false, bfrag, (short)0, cz, false, false);

#pragma unroll
    for (int g = 0; g < 8; ++g) {
      float ex = __expf(e[g]);
      nsum[g] += ex;
      wsum[g] = fmaf(ex, nue, wsum[g]);
    }
  }

  // Normalize, clamp, store. D fragment: VGPR g holds row M = g + 8*half,
  // col N = r -> each half-wave writes 64 contiguous bytes per g.
  float* __restrict__ ob = out + (size_t)b * (BSZ * BSZ);
#pragma unroll
  for (int g = 0; g < 8; ++g) {
    float denom = fmaxf(nsum[g], 1e-8f);
    float resv  = fminf(fmaxf(wsum[g] / denom, 0.0f), 1.0f);
    int m = g + 8 * half;
    ob[m * BSZ + r] = resv;
  }
}

extern "C" void kernel_launch(void* const* d_in, const int* in_sizes, int n_in,
                              void* d_out, int out_size, void* d_ws, size_t ws_size,
                              hipStream_t stream) {
  (void)n_in; (void)d_ws; (void)ws_size; (void)out_size;
  const float* params = (const float*)d_in[0];
  float* out = (float*)d_out;
  const int nBatch = in_sizes[0] / (3 * KG + 4 * KG);  // 28 floats per batch
  const int blocks = (nBatch + WAVES_PER_BLOCK - 1) / WAVES_PER_BLOCK;
  smoe_decoder_wmma<<<blocks, 32 * WAVES_PER_BLOCK, 0, stream>>>(params, out, nBatch);
}